// Grasp_PointNet2_15814069584235
// MI455X (gfx1250) — compile-verified
//
#include <hip/hip_runtime.h>
#include <hip/hip_bf16.h>
#include <math.h>

typedef __attribute__((ext_vector_type(2))) float v2f;
typedef __attribute__((ext_vector_type(8))) float v8f;

// ---------------------------------------------------------------------------
// Tiled GEMM: Y[M,N] = act(X[M,Kp] * W[K,N] + bias[N])
// X row pitch Kp (multiple of 4); columns K..Kp-1 of X must be zero.
// One wave computes a 16x64 output tile: 4 x (16x16) WMMA accumulators.
// Fast path: 1 b64 A load + 8 B loads (2 bases, immediate offsets) issued
// before the 4 v_wmma so loads overlap WMMA execution.
// act: 0 = none, 1 = relu, 2 = sigmoid
// ---------------------------------------------------------------------------
__global__ void gemm_wmma_f32(const float* __restrict__ X,
                              const float* __restrict__ W,
                              const float* __restrict__ bias,
                              float* __restrict__ Y,
                              int M, int N, int K, int Kp, int act)
{
    const int lane = threadIdx.x & 31;
    const int wave = threadIdx.x >> 5;
    const int wavesPerBlk = blockDim.x >> 5;

    const int mtiles  = (M + 15) >> 4;
    const int n4tiles = (N + 63) >> 6;            // 64-wide N macro tiles
    const long tile = (long)blockIdx.x * wavesPerBlk + wave;
    if (tile >= (long)mtiles * n4tiles) return;   // wave-uniform exit

    const int mt = (int)(tile / n4tiles);
    const int nt = (int)(tile % n4tiles);
    const int m0 = mt << 4;
    const int n0 = nt << 6;

    const int ln = lane & 15;     // A: M row ; B/C/D: N column within sub-tile
    const int hi = lane >> 4;     // selects K pair (A/B) / M+8 (C/D)

    int  ncol[4];
    bool ok[4];
    int  ncl[4];
#pragma unroll
    for (int t = 0; t < 4; ++t) {
        ncol[t] = n0 + 16 * t + ln;
        ok[t]   = (ncol[t] < N);
        ncl[t]  = ok[t] ? ncol[t] : N - 1;        // clamped (junk col never stored)
    }

    const int arow = m0 + ln;
    const long xrow = (long)(arow < M ? arow : M - 1) * Kp;
    const float* Ap = X + xrow + 2 * hi;          // 8B aligned (xrow, 2hi even)

    v8f c[4];
#pragma unroll
    for (int t = 0; t < 4; ++t) {
        const float bv = bias[ncl[t]];
#pragma unroll
        for (int r = 0; r < 8; ++r) c[t][r] = bv;
    }

    const int kmain = K & ~3;                     // unconditional K region
    const bool fullN = (n0 + 64 <= N);

    int k = 0;
    if (fullN) {
        // ---- fast path: shared W base, immediate offsets 0/64/128/192 bytes
        const float* Wr = W + (long)(2 * hi) * N + n0 + ln;   // row 2*hi, col base
        for (; k < kmain; k += 4) {
            v2f a = *(const v2f*)(Ap + k);                    // global_load_b64
            const float* w0 = Wr + (long)k * N;               // row k + 2*hi
            const float* w1 = w0 + N;                         // row k + 2*hi + 1
            // all loads first (distinct regs) -> staged loadcnt waits
            v2f b0, b1, b2, b3;
            b0[0] = w0[0];  b0[1] = w1[0];
            b1[0] = w0[16]; b1[1] = w1[16];
            b2[0] = w0[32]; b2[1] = w1[32];
            b3[0] = w0[48]; b3[1] = w1[48];
            c[0] = __builtin_amdgcn_wmma_f32_16x16x4_f32(false, a, false, b0, (short)0, c[0], false, false);
            c[1] = __builtin_amdgcn_wmma_f32_16x16x4_f32(false, a, false, b1, (short)0, c[1], false, false);
            c[2] = __builtin_amdgcn_wmma_f32_16x16x4_f32(false, a, false, b2, (short)0, c[2], false, false);
            c[3] = __builtin_amdgcn_wmma_f32_16x16x4_f32(false, a, false, b3, (short)0, c[3], false, false);
        }
    } else {
        // ---- generic path (partial N tile: heads only, tiny M)
        for (; k < kmain; k += 4) {
            v2f a = *(const v2f*)(Ap + k);
            const long r0 = (long)(k + 2 * hi) * N;
            const long r1 = r0 + N;
            v2f b0, b1, b2, b3;
            b0[0] = W[r0 + ncl[0]]; b0[1] = W[r1 + ncl[0]];
            b1[0] = W[r0 + ncl[1]]; b1[1] = W[r1 + ncl[1]];
            b2[0] = W[r0 + ncl[2]]; b2[1] = W[r1 + ncl[2]];
            b3[0] = W[r0 + ncl[3]]; b3[1] = W[r1 + ncl[3]];
            c[0] = __builtin_amdgcn_wmma_f32_16x16x4_f32(false, a, false, b0, (short)0, c[0], false, false);
            c[1] = __builtin_amdgcn_wmma_f32_16x16x4_f32(false, a, false, b1, (short)0, c[1], false, false);
            c[2] = __builtin_amdgcn_wmma_f32_16x16x4_f32(false, a, false, b2, (short)0, c[2], false, false);
            c[3] = __builtin_amdgcn_wmma_f32_16x16x4_f32(false, a, false, b3, (short)0, c[3], false, false);
        }
    }
    // ---- K tail (K % 4 != 0): A zero-padded to Kp; B clamp + value select
    for (; k < Kp; k += 4) {
        const int ka0 = k + 2 * hi;
        const int ka1 = ka0 + 1;
        const int kc0 = (ka0 < K) ? ka0 : K - 1;
        const int kc1 = (ka1 < K) ? ka1 : K - 1;
        v2f a = *(const v2f*)(Ap + k);
        v2f b[4];
#pragma unroll
        for (int t = 0; t < 4; ++t) {
            float b0 = W[(long)kc0 * N + ncl[t]]; b0 = (ka0 < K) ? b0 : 0.0f;
            float b1 = W[(long)kc1 * N + ncl[t]]; b1 = (ka1 < K) ? b1 : 0.0f;
            b[t][0] = b0; b[t][1] = b1;
        }
#pragma unroll
        for (int t = 0; t < 4; ++t) {
            c[t] = __builtin_amdgcn_wmma_f32_16x16x4_f32(
                       false, a, false, b[t], (short)0, c[t], false, false);
        }
    }

#pragma unroll
    for (int r = 0; r < 8; ++r) {
        const int row = m0 + r + 8 * hi;
        if (row < M) {
            float* yrow = Y + (long)row * N;
#pragma unroll
            for (int t = 0; t < 4; ++t) {
                float v = c[t][r];
                if (act == 1)      v = fmaxf(v, 0.0f);
                else if (act == 2) v = 1.0f / (1.0f + expf(-v));
                if (ok[t]) yrow[ncol[t]] = v;
            }
        }
    }
}

// ---------------------------------------------------------------------------
// Farthest point sampling. One block per batch. Emits idx[t] = far (pre-update)
// then dist = min(dist, d(x, x[far])), far = first-argmax(dist).
// ---------------------------------------------------------------------------
__global__ void fps_kernel(const float* __restrict__ xyz, int n, int npoint,
                           int* __restrict__ idx_out, float* __restrict__ new_xyz)
{
    __shared__ float dist[4096];
    __shared__ float red_v[256];
    __shared__ int   red_i[256];
    __shared__ float cur[3];
    __shared__ int   curfar;

    const int tid = threadIdx.x;
    const float* p = xyz + (long)blockIdx.x * n * 3;

    for (int i = tid; i < n; i += blockDim.x) dist[i] = 1e10f;
    int far = 0;
    __syncthreads();

    for (int t = 0; t < npoint; ++t) {
        if (tid == 0) {
            idx_out[blockIdx.x * npoint + t] = far;
            cur[0] = p[far * 3 + 0];
            cur[1] = p[far * 3 + 1];
            cur[2] = p[far * 3 + 2];
            float* nx = new_xyz + ((long)blockIdx.x * npoint + t) * 3;
            nx[0] = cur[0]; nx[1] = cur[1]; nx[2] = cur[2];
        }
        __syncthreads();
        const float cx = cur[0], cy = cur[1], cz = cur[2];

        float best = -1.0f;
        int   bi = 0x7fffffff;
        for (int i = tid; i < n; i += blockDim.x) {
            float dx = p[i * 3 + 0] - cx;
            float dy = p[i * 3 + 1] - cy;
            float dz = p[i * 3 + 2] - cz;
            float d = dx * dx + dy * dy + dz * dz;
            float nd = fminf(dist[i], d);
            dist[i] = nd;
            if (nd > best || (nd == best && i < bi)) { best = nd; bi = i; }
        }
        red_v[tid] = best; red_i[tid] = bi;
        __syncthreads();
        for (int s = 128; s > 0; s >>= 1) {
            if (tid < s) {
                float ov = red_v[tid + s]; int oi = red_i[tid + s];
                if (ov > red_v[tid] || (ov == red_v[tid] && oi < red_i[tid])) {
                    red_v[tid] = ov; red_i[tid] = oi;
                }
            }
            __syncthreads();
        }
        if (tid == 0) curfar = red_i[0];
        __syncthreads();
        far = curfar;
        __syncthreads();
    }
}

// ---------------------------------------------------------------------------
// Ball query + group. One wave (32 threads) per query center. First nsample
// points (ascending index) with sqr <= r2; pad with the first hit. Writes rows
// [centered xyz (3) | gathered feats (C) | zero pad] at row pitch `pitch`.
// ---------------------------------------------------------------------------
__global__ void ballquery_group(const float* __restrict__ xyz,
                                const float* __restrict__ new_xyz,
                                const float* __restrict__ feats,
                                int n, int S, int C, int nsample, float r2,
                                int pitch, float* __restrict__ Xout)
{
    const int b = blockIdx.x / S;
    const int lane = threadIdx.x;
    const float* p = xyz + (long)b * n * 3;
    const float* q = new_xyz + (long)blockIdx.x * 3;

    __shared__ int sel[64];

    const float cx = q[0], cy = q[1], cz = q[2];
    int cnt = 0;  // uniform across the wave

    for (int base = 0; base < n && cnt < nsample; base += 32) {
        int i = base + lane;
        bool in = false;
        if (i < n) {
            float dx = p[i * 3 + 0] - cx;
            float dy = p[i * 3 + 1] - cy;
            float dz = p[i * 3 + 2] - cz;
            in = (dx * dx + dy * dy + dz * dz) <= r2;
        }
        unsigned long long m = __ballot(in);
        int prefix = __popcll(m & ((1ull << lane) - 1ull));
        if (in && (cnt + prefix) < nsample) sel[cnt + prefix] = i;
        cnt += __popcll(m);
    }
    if (cnt > nsample) cnt = nsample;
    __syncthreads();

    int c0 = (cnt > 0) ? sel[0] : (n - 1);

    for (int j = lane; j < nsample; j += 32) {
        int id = (j < cnt) ? sel[j] : c0;
        float* row = Xout + ((long)blockIdx.x * nsample + j) * pitch;
        row[0] = p[id * 3 + 0] - cx;
        row[1] = p[id * 3 + 1] - cy;
        row[2] = p[id * 3 + 2] - cz;
        if (C > 0) {
            const float* f = feats + ((long)b * n + id) * C;
            for (int cc = 0; cc < C; ++cc) row[3 + cc] = f[cc];
        }
        for (int cc = 3 + C; cc < pitch; ++cc) row[cc] = 0.0f;   // K pad
    }
}

// ---------------------------------------------------------------------------
// Max pool over the group dimension: in[G, gsize, C] -> out[G, C]
// ---------------------------------------------------------------------------
__global__ void maxpool_kernel(const float* __restrict__ in, float* __restrict__ out,
                               int G, int gsize, int C)
{
    long idx = (long)blockIdx.x * blockDim.x + threadIdx.x;
    if (idx >= (long)G * C) return;
    long g = idx / C;
    int  c = (int)(idx % C);
    const float* base = in + g * gsize * (long)C + c;
    float m = base[0];
    for (int j = 1; j < gsize; ++j) m = fmaxf(m, base[(long)j * C]);
    out[idx] = m;
}

// ---------------------------------------------------------------------------
// X3[rows, 260] = concat(l2_xyz[rows,3], l2_feats[rows,256], zero-pad(1))
// ---------------------------------------------------------------------------
__global__ void concat3_kernel(const float* __restrict__ l2xyz,
                               const float* __restrict__ l2f,
                               float* __restrict__ X, int rows)
{
    long idx = (long)blockIdx.x * blockDim.x + threadIdx.x;
    if (idx >= (long)rows * 260) return;
    long r = idx / 260;
    int  c = (int)(idx % 260);
    float v;
    if (c < 3)        v = l2xyz[r * 3 + c];
    else if (c < 259) v = l2f[r * 256 + (c - 3)];
    else              v = 0.0f;
    X[idx] = v;
}

// ---------------------------------------------------------------------------
// Final head: out[B,20] = [sigmoid-label(2) | pl(3) | rl(6) | pr(3) | rr(6)]
// ---------------------------------------------------------------------------
__global__ void heads_kernel(const float* __restrict__ lab,
                             const float* __restrict__ res,
                             const float* __restrict__ pm, const float* __restrict__ ps,
                             const float* __restrict__ rm, const float* __restrict__ rs,
                             float* __restrict__ out)
{
    int b = blockIdx.x;
    int t = threadIdx.x;
    if (t >= 20) return;
    float v;
    if (t < 2)       v = lab[b * 2 + t];
    else if (t < 5)  v = res[b * 18 + (t - 2)]  * ps[t - 2]  + pm[t - 2];
    else if (t < 11) v = res[b * 18 + (t - 2)]  * rs[t - 5]  + rm[t - 5];
    else if (t < 14) v = res[b * 18 + (t - 2)]  * ps[t - 11] + pm[t - 11];
    else             v = res[b * 18 + (t - 2)]  * rs[t - 14] + rm[t - 14];
    out[b * 20 + t] = v;
}

// ---------------------------------------------------------------------------
extern "C" void kernel_launch(void* const* d_in, const int* in_sizes, int n_in,
                              void* d_out, int out_size, void* d_ws, size_t ws_size,
                              hipStream_t stream)
{
    (void)in_sizes; (void)n_in; (void)out_size; (void)ws_size;
    const int B = 16, N = 4096;

    const float* points = (const float*)d_in[0];
    const float* w10 = (const float*)d_in[1];  const float* b10 = (const float*)d_in[2];
    const float* w11 = (const float*)d_in[3];  const float* b11 = (const float*)d_in[4];
    const float* w12 = (const float*)d_in[5];  const float* b12 = (const float*)d_in[6];
    const float* w20 = (const float*)d_in[7];  const float* b20 = (const float*)d_in[8];
    const float* w21 = (const float*)d_in[9];  const float* b21 = (const float*)d_in[10];
    const float* w22 = (const float*)d_in[11]; const float* b22 = (const float*)d_in[12];
    const float* w30 = (const float*)d_in[13]; const float* b30 = (const float*)d_in[14];
    const float* w31 = (const float*)d_in[15]; const float* b31 = (const float*)d_in[16];
    const float* w32 = (const float*)d_in[17]; const float* b32 = (const float*)d_in[18];
    const float* lw1 = (const float*)d_in[19]; const float* lb1 = (const float*)d_in[20];
    const float* lw2 = (const float*)d_in[21]; const float* lb2 = (const float*)d_in[22];
    const float* fw1 = (const float*)d_in[23]; const float* fb1 = (const float*)d_in[24];
    const float* fw2 = (const float*)d_in[25]; const float* fb2 = (const float*)d_in[26];
    const float* pos_mean = (const float*)d_in[27];
    const float* pos_std  = (const float*)d_in[28];
    const float* rot_mean = (const float*)d_in[29];
    const float* rot_std  = (const float*)d_in[30];
    float* out = (float*)d_out;

    // ---- workspace layout (bump allocator, 256B aligned) ----
    char* ws = (char*)d_ws;
    size_t off = 0;
    auto alloc = [&](size_t bytes) -> char* {
        char* r = ws + off;
        off = (off + bytes + 255) & ~(size_t)255;
        return r;
    };
    int*   fps1_idx = (int*)  alloc((size_t)B * 512 * 4);
    float* l1_xyz   = (float*)alloc((size_t)B * 512 * 3 * 4);
    float* l1_feats = (float*)alloc((size_t)B * 512 * 128 * 4);
    int*   fps2_idx = (int*)  alloc((size_t)B * 128 * 4);
    float* l2_xyz   = (float*)alloc((size_t)B * 128 * 3 * 4);
    float* l2_feats = (float*)alloc((size_t)B * 128 * 256 * 4);
    float* gvec     = (float*)alloc((size_t)B * 1024 * 4);
    float* labh     = (float*)alloc((size_t)B * 256 * 4);
    float* lab      = (float*)alloc((size_t)B * 2 * 4);
    float* resh     = (float*)alloc((size_t)B * 512 * 4);
    float* res      = (float*)alloc((size_t)B * 18 * 4);
    float* bufA     = (float*)alloc((size_t)262144 * 132 * 4);   // ping
    float* bufB     = (float*)alloc((size_t)262144 * 132 * 4);   // pong

    auto gemm = [&](const float* X, const float* W, const float* Bs, float* Y,
                    int M, int Nn, int K, int Kp, int act) {
        long tiles = (long)((M + 15) / 16) * ((Nn + 63) / 64);
        int wavesPerBlk = 8;                                     // 256 threads
        unsigned blocks = (unsigned)((tiles + wavesPerBlk - 1) / wavesPerBlk);
        gemm_wmma_f32<<<blocks, 256, 0, stream>>>(X, W, Bs, Y, M, Nn, K, Kp, act);
    };

    // ---------------- SA1: npoint=512, r=0.2, ns=32, C: 3 -> 64 -> 64 -> 128
    fps_kernel<<<B, 256, 0, stream>>>(points, N, 512, fps1_idx, l1_xyz);
    ballquery_group<<<B * 512, 32, 0, stream>>>(points, l1_xyz, nullptr,
                                                N, 512, 0, 32, 0.04f, 4, bufA);
    gemm(bufA, w10, b10, bufB, 262144, 64, 3, 4, 1);
    gemm(bufB, w11, b11, bufA, 262144, 64, 64, 64, 1);
    gemm(bufA, w12, b12, bufB, 262144, 128, 64, 64, 1);
    maxpool_kernel<<<(8192 * 128 + 255) / 256, 256, 0, stream>>>(bufB, l1_feats, 8192, 32, 128);

    // ---------------- SA2: npoint=128, r=0.4, ns=64, C: 131 -> 128 -> 128 -> 256
    fps_kernel<<<B, 256, 0, stream>>>(l1_xyz, 512, 128, fps2_idx, l2_xyz);
    ballquery_group<<<B * 128, 32, 0, stream>>>(l1_xyz, l2_xyz, l1_feats,
                                                512, 128, 128, 64, 0.16f, 132, bufA);
    gemm(bufA, w20, b20, bufB, 131072, 128, 131, 132, 1);
    gemm(bufB, w21, b21, bufA, 131072, 128, 128, 128, 1);
    gemm(bufA, w22, b22, bufB, 131072, 256, 128, 128, 1);
    maxpool_kernel<<<(2048 * 256 + 255) / 256, 256, 0, stream>>>(bufB, l2_feats, 2048, 64, 256);

    // ---------------- SA3 (group_all): 259 -> 256 -> 512 -> 1024, max over 128
    concat3_kernel<<<((2048 * 260) + 255) / 256, 256, 0, stream>>>(l2_xyz, l2_feats, bufA, 2048);
    gemm(bufA, w30, b30, bufB, 2048, 256, 259, 260, 1);
    gemm(bufB, w31, b31, bufA, 2048, 512, 256, 256, 1);
    gemm(bufA, w32, b32, bufB, 2048, 1024, 512, 512, 1);
    maxpool_kernel<<<(B * 1024 + 255) / 256, 256, 0, stream>>>(bufB, gvec, B, 128, 1024);

    // ---------------- heads
    gemm(gvec, lw1, lb1, labh, B, 256, 1024, 1024, 1);
    gemm(labh, lw2, lb2, lab,  B, 2,   256,  256,  2);   // sigmoid
    gemm(gvec, fw1, fb1, resh, B, 512, 1024, 1024, 1);
    gemm(resh, fw2, fb2, res,  B, 18,  512,  512,  0);

    heads_kernel<<<B, 32, 0, stream>>>(lab, res, pos_mean, pos_std, rot_mean, rot_std, out);
}